// BatchedSplineReconstruction_86577950752949
// MI455X (gfx1250) — compile-verified
//
#include <hip/hip_runtime.h>
#include <math.h>

// Problem constants (from setup_inputs): B=2048, n_cp=32, num_points=2048, degree=3
#define NPTS   2048
#define BATCH  2048
#define NCP    32
#define MTOT   (BATCH * 2)          // rows m = b*2 + d
#define NKNOT  (NCP + 4)            // 36 knots for p=3

typedef __attribute__((ext_vector_type(2))) float v2f;
typedef __attribute__((ext_vector_type(8))) float v8f;

// ---------------------------------------------------------------------------
// Kernel 1: precompute B-spline basis (Cox–de Boor, p=3) and Bernstein basis
// basis[i*NPTS + n]  = N_i(t_n)            (32 x 2048, row major)
// bernT[i*NPTS + n]  = bern(t_n, i)        (transposed Bernstein, 32 x 2048)
// ---------------------------------------------------------------------------
__global__ void __launch_bounds__(256)
basis_precompute_kernel(float* __restrict__ basis, float* __restrict__ bernT) {
    const int n = blockIdx.x * blockDim.x + threadIdx.x;
    if (n >= NPTS) return;
    const float t = (float)n * (1.0f / 2047.0f);

    // knot vector: [0,0,0,0, 1/29 .. 28/29, 1,1,1,1]
    float knots[NKNOT];
#pragma unroll
    for (int j = 0; j < 4; ++j) { knots[j] = 0.0f; knots[NCP + j] = 1.0f; }
#pragma unroll
    for (int j = 1; j <= 28; ++j) knots[3 + j] = (float)j * (1.0f / 29.0f);

    // degree-0 basis
    float Nv[NCP + 1];
#pragma unroll
    for (int i = 0; i < NCP; ++i)
        Nv[i] = (t >= knots[i] && t < knots[i + 1]) ? 1.0f : 0.0f;
    Nv[NCP - 1] = (t >= knots[NCP - 1] && t <= knots[NCP]) ? 1.0f : 0.0f;
    Nv[NCP] = 0.0f;

    // Cox–de Boor recursion up to degree 3
#pragma unroll
    for (int d = 1; d <= 3; ++d) {
        float Nn[NCP];
#pragma unroll
        for (int i = 0; i < NCP; ++i) {
            const float d1 = knots[i + d]     - knots[i];
            const float d2 = knots[i + d + 1] - knots[i + 1];
            const float t1 = (d1 != 0.0f) ? (t - knots[i]) / d1 * Nv[i] : 0.0f;
            const float t2 = (d2 != 0.0f) ? (knots[i + d + 1] - t) / d2 * Nv[i + 1] : 0.0f;
            Nn[i] = t1 + t2;
        }
#pragma unroll
        for (int i = 0; i < NCP; ++i) Nv[i] = Nn[i];
        Nv[NCP] = 0.0f;
    }
#pragma unroll
    for (int i = 0; i < NCP; ++i)
        basis[i * NPTS + n] = Nv[i];

    // Bernstein basis, n_bez = 31 (match reference's lgamma formulation)
    const float lgN = lgammaf(32.0f);
#pragma unroll
    for (int i = 0; i < NCP; ++i) {
        const float fi = (float)i;
        const float c  = expf(lgN - lgammaf(fi + 1.0f) - lgammaf(32.0f - fi));
        const float v  = c * powf(t, fi) * powf(1.0f - t, 31.0f - fi);
        bernT[i * NPTS + n] = v;
    }
}

// ---------------------------------------------------------------------------
// WMMA helper: 16x16 fp32 output tile = A(16x32) x B(32x16) via 8 chained
// V_WMMA_F32_16X16X4_F32 (K=4 each).
// ---------------------------------------------------------------------------
__device__ __forceinline__ v8f wmma_chain_k32(const v2f a[8], const v2f b[8]) {
    v8f acc = {};
#pragma unroll
    for (int k = 0; k < 8; ++k) {
        acc = __builtin_amdgcn_wmma_f32_16x16x4_f32(
            /*neg_a=*/false, a[k], /*neg_b=*/false, b[k],
            /*c_mod=*/(short)0, acc, /*reuse_a=*/false, /*reuse_b=*/false);
    }
    return acc;
}

// ---------------------------------------------------------------------------
// Kernel 2: one wave32 per 16x16 output tile. Block = 8 waves => 16(M) x 128(N)
// slab. Computes all three curves + NURBS denominator with 4 WMMA chains.
//
// A-fragment layout (16x4 f32): lanes 0-15 -> M=lane, K={k, k+1};
//                               lanes 16-31 -> M=lane-16, K={k+2, k+3}.
// B-fragment layout (4x16 f32): lanes 0-15 -> N=lane, K rows {k, k+1};
//                               lanes 16-31 -> N=lane-16, K rows {k+2, k+3}.
// C/D layout (16x16 f32): VGPR r, lanes 0-15 -> (M=r, N=lane);
//                                  lanes 16-31 -> (M=r+8, N=lane-16).
// ---------------------------------------------------------------------------
__global__ void __launch_bounds__(256)
spline_wmma_kernel(const float* __restrict__ bspline_cp,   // [B][32][2]
                   const float* __restrict__ nurbs_cp,     // [B][32][2]
                   const float* __restrict__ w,            // [B][32]
                   const float* __restrict__ bezier_cp,    // [B][32][2]
                   const float* __restrict__ basis,        // [32][NPTS]
                   const float* __restrict__ bernT,        // [32][NPTS]
                   float* __restrict__ out) {
    const int lane = threadIdx.x & 31;
    const int wv   = threadIdx.x >> 5;
    const int n0   = (blockIdx.x * 8 + wv) * 16;   // N tile origin
    const int m0   = blockIdx.y * 16;              // M tile origin (m = b*2+d)

    const int kHi  = lane >> 4;                    // 0: lanes 0-15, 1: lanes 16-31
    const int nCol = n0 + (lane & 15);
    const int rowM = m0 + (lane & 15);
    const int bb   = rowM >> 1;                    // batch index
    const int dd   = rowM & 1;                     // coordinate (x/y)

    // ---- load B fragments (basis + bernstein), 8 x float2 each -------------
    v2f bB[8], bZ[8];
#pragma unroll
    for (int k = 0; k < 8; ++k) {
        const int kb = 4 * k + kHi * 2;
        bB[k].x = basis[(kb + 0) * NPTS + nCol];
        bB[k].y = basis[(kb + 1) * NPTS + nCol];
        bZ[k].x = bernT[(kb + 0) * NPTS + nCol];
        bZ[k].y = bernT[(kb + 1) * NPTS + nCol];
    }

    v2f a[8];

    // ---- bspline: A[m,i] = bspline_cp[b, i, d] -----------------------------
#pragma unroll
    for (int k = 0; k < 8; ++k) {
        const int kb = 4 * k + kHi * 2;
        a[k].x = bspline_cp[bb * (NCP * 2) + (kb + 0) * 2 + dd];
        a[k].y = bspline_cp[bb * (NCP * 2) + (kb + 1) * 2 + dd];
    }
    const v8f accB = wmma_chain_k32(a, bB);

    // ---- nurbs numerator: A[m,i] = w[b,i] * nurbs_cp[b, i, d] --------------
#pragma unroll
    for (int k = 0; k < 8; ++k) {
        const int kb = 4 * k + kHi * 2;
        a[k].x = w[bb * NCP + kb + 0] * nurbs_cp[bb * (NCP * 2) + (kb + 0) * 2 + dd];
        a[k].y = w[bb * NCP + kb + 1] * nurbs_cp[bb * (NCP * 2) + (kb + 1) * 2 + dd];
    }
    const v8f accN = wmma_chain_k32(a, bB);

    // ---- nurbs denominator: A[m,i] = w[b,i] (rows duplicated per d) --------
#pragma unroll
    for (int k = 0; k < 8; ++k) {
        const int kb = 4 * k + kHi * 2;
        a[k].x = w[bb * NCP + kb + 0];
        a[k].y = w[bb * NCP + kb + 1];
    }
    const v8f accD = wmma_chain_k32(a, bB);

    // ---- bezier: A[m,i] = bezier_cp[b, i, d], B = bernT --------------------
#pragma unroll
    for (int k = 0; k < 8; ++k) {
        const int kb = 4 * k + kHi * 2;
        a[k].x = bezier_cp[bb * (NCP * 2) + (kb + 0) * 2 + dd];
        a[k].y = bezier_cp[bb * (NCP * 2) + (kb + 1) * 2 + dd];
    }
    const v8f accZ = wmma_chain_k32(a, bZ);

    // ---- stores: out = [bspline | nurbs | bezier], each [B][2][NPTS] -------
    float* __restrict__ bsp = out;
    float* __restrict__ nur = out + (size_t)MTOT * NPTS;
    float* __restrict__ bez = out + 2 * (size_t)MTOT * NPTS;
    const int col = nCol;
#pragma unroll
    for (int r = 0; r < 8; ++r) {
        const int row = m0 + r + kHi * 8;
        const size_t off = (size_t)row * NPTS + col;
        __builtin_nontemporal_store(accB[r], bsp + off);
        __builtin_nontemporal_store(accN[r] / (accD[r] + 1e-8f), nur + off);
        __builtin_nontemporal_store(accZ[r], bez + off);
    }
}

// ---------------------------------------------------------------------------
extern "C" void kernel_launch(void* const* d_in, const int* in_sizes, int n_in,
                              void* d_out, int out_size, void* d_ws, size_t ws_size,
                              hipStream_t stream) {
    const float* bspline_cp = (const float*)d_in[0];   // [2048][32][2]
    const float* nurbs_cp   = (const float*)d_in[1];   // [2048][32][2]
    const float* nurbs_w    = (const float*)d_in[2];   // [2048][32]
    const float* bezier_cp  = (const float*)d_in[3];   // [2048][32][2]
    // d_in[4] = num_points (2048, fixed by setup)

    float* basis = (float*)d_ws;                 // 32*2048 floats = 256 KB
    float* bernT = basis + (size_t)NCP * NPTS;   // 32*2048 floats = 256 KB

    // Kernel 1: tiny precompute of the two shared basis matrices.
    basis_precompute_kernel<<<dim3(NPTS / 256), dim3(256), 0, stream>>>(basis, bernT);

    // Kernel 2: WMMA GEMMs. Grid: x = 2048/(8 waves * 16) = 16 N-slabs,
    //                             y = 4096/16 = 256 M-tiles.
    spline_wmma_kernel<<<dim3(16, 256), dim3(256), 0, stream>>>(
        bspline_cp, nurbs_cp, nurbs_w, bezier_cp, basis, bernT, (float*)d_out);
}